// EEGGraphAttentionLayer_62234076119834
// MI455X (gfx1250) — compile-verified
//
#include <hip/hip_runtime.h>

#define N_NODES 8192
#define F_IN    128
#define F_OUT   64
#define NEG_SLOPE 0.2f
#define BIG_NEG (-1.0e12f)

typedef __attribute__((ext_vector_type(2))) float v2f;
typedef __attribute__((ext_vector_type(4))) float v4f;
typedef __attribute__((ext_vector_type(8))) float v8f;

// ---------------------------------------------------------------------------
// Kernel 0: ws = w @ a_src, wd = w @ a_dst   (128 floats each; trivial)
// ---------------------------------------------------------------------------
__global__ void prep_wsd(const float* __restrict__ w, const float* __restrict__ a,
                         float* __restrict__ wsd) {
  int k = threadIdx.x;  // 0..127
  float s = 0.f, d = 0.f;
  #pragma unroll
  for (int f = 0; f < F_OUT; ++f) {
    float wv = w[k * F_OUT + f];
    s += wv * a[f];
    d += wv * a[F_OUT + f];
  }
  wsd[k] = s;            // ws
  wsd[F_IN + k] = d;     // wd
}

// ---------------------------------------------------------------------------
// Kernel 1: [s | d] = h(8192x128) @ B(128x16) with B col0=ws, col1=wd.
// One wave per 16-row tile of h; 32 steps of V_WMMA_F32_16X16X4_F32.
// A 16x4 f32 layout: lanes 0-15 -> M=lane, VGPR0/1 = K{0,1}; lanes 16-31 K{2,3}.
// B 4x16 f32 layout (mirror): lanes 0-15 -> N=lane, VGPR0/1 = K{0,1}; hi lanes K{2,3}.
// C/D 16x16 f32: lanes 0-15 -> N=lane, VGPR r = M=r; lanes 16-31 -> M=8+r.
// ---------------------------------------------------------------------------
__global__ __launch_bounds__(32) void wmma_sd(const float* __restrict__ h,
                                              const float* __restrict__ wsd,
                                              float* __restrict__ sbuf,
                                              float* __restrict__ dbuf) {
  __shared__ float B[2 * F_IN];
  const int lane = threadIdx.x;
  for (int i = lane; i < 2 * F_IN; i += 32) B[i] = wsd[i];
  __syncthreads();

  const int row0 = blockIdx.x * 16;
  const int m    = lane & 15;       // M for A, N for B/C
  const int grp  = lane >> 4;       // selects K pair
  const int koff = grp * 2;
  const float* hrow = h + (size_t)(row0 + m) * F_IN + koff;

  const int col  = m;               // output column (0 -> s, 1 -> d)
  const int cidx = (col < 2) ? col : 0;
  const bool live = (col < 2);

  v8f c = {};
  #pragma unroll
  for (int kk = 0; kk < F_IN / 4; ++kk) {
    const int k0 = kk * 4;
    v2f av = *(const v2f*)(hrow + k0);           // global_load_b64
    float b0 = B[cidx * F_IN + k0 + koff];       // ds_load
    float b1 = B[cidx * F_IN + k0 + koff + 1];
    v2f bv;
    bv.x = live ? b0 : 0.f;
    bv.y = live ? b1 : 0.f;
    c = __builtin_amdgcn_wmma_f32_16x16x4_f32(
        /*neg_a=*/false, av, /*neg_b=*/false, bv,
        /*c_mod=*/(short)0, c, /*reuse_a=*/false, /*reuse_b=*/false);
  }

  // lanes 0/16 hold s (N=0) for M = grp*8 + r ; lanes 1/17 hold d (N=1)
  if (col == 0) {
    #pragma unroll
    for (int r = 0; r < 8; ++r) sbuf[row0 + grp * 8 + r] = c[r];
  } else if (col == 1) {
    #pragma unroll
    for (int r = 0; r < 8; ++r) dbuf[row0 + grp * 8 + r] = c[r];
  }
}

// ---------------------------------------------------------------------------
// Kernel 2: fused mask + leaky_relu + row softmax. One block per row.
// ---------------------------------------------------------------------------
__device__ __forceinline__ float wave_max(float v) {
  #pragma unroll
  for (int o = 16; o > 0; o >>= 1) v = fmaxf(v, __shfl_xor(v, o, 32));
  return v;
}
__device__ __forceinline__ float wave_sum(float v) {
  #pragma unroll
  for (int o = 16; o > 0; o >>= 1) v += __shfl_xor(v, o, 32);
  return v;
}

__global__ __launch_bounds__(1024) void attn_softmax(const float* __restrict__ adj,
                                                     const float* __restrict__ sbuf,
                                                     const float* __restrict__ dbuf,
                                                     float* __restrict__ out) {
  __shared__ float ldsd[N_NODES];   // 32 KB: d[] cached per block
  __shared__ float red[32];

  const int tid = threadIdx.x;
  const int row = blockIdx.x;
  const int lane = tid & 31;
  const int wid  = tid >> 5;

  // stage d[] into LDS (float4 loads)
  {
    v4f* l4 = (v4f*)ldsd;
    const v4f* d4 = (const v4f*)dbuf;
    #pragma unroll
    for (int i = tid; i < N_NODES / 4; i += 1024) l4[i] = d4[i];
  }
  __syncthreads();

  const float si = sbuf[row];
  const v4f* arow = (const v4f*)(adj + (size_t)row * N_NODES);

  float att[8];
  float mx = BIG_NEG;
  #pragma unroll
  for (int half = 0; half < 2; ++half) {
    const int f4i = tid + half * 1024;
    v4f av = __builtin_nontemporal_load(arow + f4i);  // streamed once: NT
    const int j0 = f4i * 4;
    #pragma unroll
    for (int cc = 0; cc < 4; ++cc) {
      float x = si + ldsd[j0 + cc];
      float e = (x > 0.f) ? x : NEG_SLOPE * x;        // leaky_relu
      float at = (av[cc] > 0.f) ? e : BIG_NEG;        // adjacency mask
      att[half * 4 + cc] = at;
      mx = fmaxf(mx, at);
    }
  }

  // block max
  mx = wave_max(mx);
  if (lane == 0) red[wid] = mx;
  __syncthreads();
  if (wid == 0) {
    float t = wave_max(red[lane]);
    if (lane == 0) red[0] = t;
  }
  __syncthreads();
  const float M = red[0];
  __syncthreads();

  // exp + block sum
  float lsum = 0.f;
  #pragma unroll
  for (int k = 0; k < 8; ++k) {
    att[k] = __expf(att[k] - M);
    lsum += att[k];
  }
  lsum = wave_sum(lsum);
  if (lane == 0) red[wid] = lsum;
  __syncthreads();
  if (wid == 0) {
    float t = wave_sum(red[lane]);
    if (lane == 0) red[0] = t;
  }
  __syncthreads();
  const float inv = 1.f / red[0];

  // normalized NT stores (write-once stream)
  v4f* orow = (v4f*)(out + (size_t)row * N_NODES);
  #pragma unroll
  for (int half = 0; half < 2; ++half) {
    const int f4i = tid + half * 1024;
    v4f ov;
    ov.x = att[half * 4 + 0] * inv;
    ov.y = att[half * 4 + 1] * inv;
    ov.z = att[half * 4 + 2] * inv;
    ov.w = att[half * 4 + 3] * inv;
    __builtin_nontemporal_store(ov, orow + f4i);
  }
}

// ---------------------------------------------------------------------------
// Launch. Workspace layout (floats): [0,128)=ws, [128,256)=wd,
// [256,256+8192)=s, [256+8192, 256+16384)=d.   (~66.5 KB)
// ---------------------------------------------------------------------------
extern "C" void kernel_launch(void* const* d_in, const int* in_sizes, int n_in,
                              void* d_out, int out_size, void* d_ws, size_t ws_size,
                              hipStream_t stream) {
  const float* h   = (const float*)d_in[0];
  const float* adj = (const float*)d_in[1];
  const float* w   = (const float*)d_in[2];
  const float* a   = (const float*)d_in[3];

  float* wsd  = (float*)d_ws;            // ws | wd (256 floats)
  float* sbuf = wsd + 2 * F_IN;          // 8192 floats
  float* dbuf = sbuf + N_NODES;          // 8192 floats
  float* out  = (float*)d_out;

  prep_wsd<<<1, 128, 0, stream>>>(w, a, wsd);
  wmma_sd<<<N_NODES / 16, 32, 0, stream>>>(h, wsd, sbuf, dbuf);
  attn_softmax<<<N_NODES, 1024, 0, stream>>>(adj, sbuf, dbuf, out);
}